// GATNetHeadsChanged3LayersEmbeddingDim256Entropy_43739946943266
// MI455X (gfx1250) — compile-verified
//
#include <hip/hip_runtime.h>
#include <hip/hip_bf16.h>
#include <math.h>

// ---------------------------------------------------------------------------
// GATNet (2-head GATConv, 3-layer MLP head) + full pairwise cdist, MI455X.
//
// Cost model (MI455X, 23.3 TB/s HBM, 192 MB L2):
//   * dist output = 16384^2 * 4B = 1.07 GB streaming stores -> ~46us floor.
//   * edge aggregation ~0.55 GB gather + ~0.55 GB f32 atomics, both L2-resident
//     (H = 16 MB, agg = 16 MB).
//   * all GEMMs < 5.5 GFLOP total -> negligible; done in f32 WMMA
//     (v_wmma_f32_16x16x4_f32) to match reference precision.
// ---------------------------------------------------------------------------

typedef float v2f __attribute__((ext_vector_type(2)));
typedef float v8f __attribute__((ext_vector_type(8)));
typedef float f4  __attribute__((ext_vector_type(4)));

#define Nn 16384
#define Ee 524288
#define EP (Ee + Nn)   // edges + self loops = 540672

// -------- monotonic float<->uint transform for atomicMax on floats ----------
__device__ __forceinline__ unsigned orderedU(float f) {
    unsigned u = __float_as_uint(f);
    return (u & 0x80000000u) ? ~u : (u | 0x80000000u);
}
__device__ __forceinline__ float fromOrderedU(unsigned u) {
    u = (u & 0x80000000u) ? (u & 0x7fffffffu) : ~u;
    return __uint_as_float(u);
}

// ---------------------------- init ------------------------------------------
__global__ void k_init(float* __restrict__ agg, unsigned* __restrict__ m_u,
                       float* __restrict__ denom) {
    int i = blockIdx.x * blockDim.x + threadIdx.x;
    if (i < Nn * 256) agg[i] = 0.f;
    if (i < Nn * 2) { m_u[i] = 0u; denom[i] = 0.f; }
}

// ------------------- f32 WMMA GEMM: C[M,NC] = act(A[M,K]) @ B[K,NC] ---------
// One 16x16 output tile per wave32.  ISA 7.12.2 layouts:
//   A 16x4 :  v_j, lane l : row = l&15, K = k + 2*(l>>4) + j
//   B 4x16 :  v_j, lane l : col = l&15 (+nothing), K = k + 2*(l>>4) + j  (row-striped)
//   C 16x16:  v_r, lane l : row = r + 8*(l>>4), col = l&15
template <bool RELU_IN, bool RELU_OUT, bool IN_BIAS, bool OUT_BIAS>
__global__ void k_wmma_gemm(const float* __restrict__ A, const float* __restrict__ inBias,
                            const float* __restrict__ B, const float* __restrict__ outBias,
                            float* __restrict__ C, int M, int K, int NC) {
    const int lane = threadIdx.x & 31;
    const int wave = blockIdx.x * (blockDim.x >> 5) + (threadIdx.x >> 5);
    const int ntiles = NC >> 4;
    const int tm = wave / ntiles, tn = wave % ntiles;
    if (tm * 16 >= M) return;
    const int half = lane >> 4, lr = lane & 15;
    const int arow = tm * 16 + lr;
    const int bcol = tn * 16 + lr;

    v8f c = {};
    for (int k = 0; k < K; k += 4) {
        const int ka = k + 2 * half;
        float a0 = A[(size_t)arow * K + ka];
        float a1 = A[(size_t)arow * K + ka + 1];
        if (RELU_IN) {
            float bb0 = IN_BIAS ? inBias[ka]     : 0.f;
            float bb1 = IN_BIAS ? inBias[ka + 1] : 0.f;
            a0 = fmaxf(a0 + bb0, 0.f);
            a1 = fmaxf(a1 + bb1, 0.f);
        }
        v2f av = {a0, a1};
        v2f bv = {B[(size_t)ka * NC + bcol], B[(size_t)(ka + 1) * NC + bcol]};
        c = __builtin_amdgcn_wmma_f32_16x16x4_f32(false, av, false, bv,
                                                  (short)0, c, false, false);
    }
#pragma unroll
    for (int r = 0; r < 8; ++r) {
        const int row = tm * 16 + r + 8 * half;
        float v = c[r];
        if (OUT_BIAS) v += outBias[bcol];
        if (RELU_OUT) v = fmaxf(v, 0.f);
        C[(size_t)row * NC + bcol] = v;
    }
}

// ------------------- attention scores a_s, a_d ------------------------------
__global__ void k_attn_scores(const float* __restrict__ H, const float* __restrict__ att_src,
                              const float* __restrict__ att_dst,
                              float* __restrict__ a_s, float* __restrict__ a_d) {
    int t = blockIdx.x * blockDim.x + threadIdx.x;   // (node, head)
    if (t >= Nn * 2) return;
    int n = t >> 1, h = t & 1;
    const f4* hp = (const f4*)(H + (size_t)n * 256 + h * 128);
    const f4* as = (const f4*)(att_src + h * 128);
    const f4* ad = (const f4*)(att_dst + h * 128);
    float ss = 0.f, sd = 0.f;
#pragma unroll 8
    for (int i = 0; i < 32; ++i) {
        f4 hv = hp[i], av = as[i], dv = ad[i];
        ss += hv.x * av.x + hv.y * av.y + hv.z * av.z + hv.w * av.w;
        sd += hv.x * dv.x + hv.y * dv.y + hv.z * dv.z + hv.w * dv.w;
    }
    a_s[t] = ss; a_d[t] = sd;
}

// ------------------- edge pass 1: leaky-relu score + segment max ------------
__global__ void k_edge1(const long long* __restrict__ ei_in, const float* __restrict__ a_s,
                        const float* __restrict__ a_d, long long* __restrict__ ei_out,
                        float* __restrict__ alpha, unsigned* __restrict__ m_u) {
    int e = blockIdx.x * blockDim.x + threadIdx.x;
    if (e >= EP) return;
    int s, d;
    if (e < Ee) { s = (int)ei_in[e]; d = (int)ei_in[Ee + e]; }
    else        { s = d = e - Ee; }
    ei_out[e] = (long long)s;
    ei_out[EP + e] = (long long)d;
#pragma unroll
    for (int h = 0; h < 2; ++h) {
        float v = a_s[s * 2 + h] + a_d[d * 2 + h];
        v = (v > 0.f) ? v : 0.2f * v;                 // leaky_relu(., 0.2)
        alpha[e * 2 + h] = v;                          // stage raw score in alpha slot
        atomicMax(&m_u[d * 2 + h], orderedU(v));
    }
}

// ------------------- edge pass 2: exp(e - m) + segment sum ------------------
__global__ void k_edge2(const long long* __restrict__ ei_out, const unsigned* __restrict__ m_u,
                        float* __restrict__ alpha, float* __restrict__ denom) {
    int e = blockIdx.x * blockDim.x + threadIdx.x;
    if (e >= EP) return;
    int d = (int)ei_out[EP + e];
#pragma unroll
    for (int h = 0; h < 2; ++h) {
        float v = alpha[e * 2 + h];
        float a = expf(v - fromOrderedU(m_u[d * 2 + h]));
        alpha[e * 2 + h] = a;                          // unnormalized alpha
        atomicAdd(&denom[d * 2 + h], a);
    }
}

// ------------------- edge pass 3: normalize + message aggregation -----------
// One wave32 per edge: lane covers 8 contiguous channels (lanes 0-15 head 0,
// lanes 16-31 head 1); 8 global_atomic_add_f32 per lane into agg[dst].
__global__ void k_edge3(const long long* __restrict__ ei_out, const float* __restrict__ denom,
                        const float* __restrict__ H, float* __restrict__ alpha,
                        float* __restrict__ agg) {
    int wave = blockIdx.x * (blockDim.x >> 5) + (threadIdx.x >> 5);
    int lane = threadIdx.x & 31;
    if (wave >= EP) return;
    int s = (int)ei_out[wave];
    int d = (int)ei_out[EP + wave];
    int head = lane >> 4;
    float a = alpha[wave * 2 + head] / (denom[d * 2 + head] + 1e-16f);
    if ((lane & 15) == 0) alpha[wave * 2 + head] = a;  // final normalized alpha -> output
    const f4* hs = (const f4*)(H + (size_t)s * 256);
    float* ag = agg + (size_t)d * 256;
    int c0 = lane * 8;
    f4 h0 = hs[lane * 2 + 0];
    f4 h1 = hs[lane * 2 + 1];
    atomicAdd(&ag[c0 + 0], h0.x * a);
    atomicAdd(&ag[c0 + 1], h0.y * a);
    atomicAdd(&ag[c0 + 2], h0.z * a);
    atomicAdd(&ag[c0 + 3], h0.w * a);
    atomicAdd(&ag[c0 + 4], h1.x * a);
    atomicAdd(&ag[c0 + 5], h1.y * a);
    atomicAdd(&ag[c0 + 6], h1.z * a);
    atomicAdd(&ag[c0 + 7], h1.w * a);
}

// ------------------- final tiny GEMM: y = h3 @ W2 + b2, plus |y|^2 ----------
__global__ void k_y(const float* __restrict__ h3, const float* __restrict__ W2,
                    const float* __restrict__ b2, f4* __restrict__ yp) {
    int n = blockIdx.x * blockDim.x + threadIdx.x;
    if (n >= Nn) return;
    float y0 = b2[0], y1 = b2[1], y2 = b2[2];
    const float* hr = h3 + (size_t)n * 64;
#pragma unroll 8
    for (int k = 0; k < 64; ++k) {
        float hv = hr[k];
        y0 += hv * W2[k * 3 + 0];
        y1 += hv * W2[k * 3 + 1];
        y2 += hv * W2[k * 3 + 2];
    }
    f4 o; o.x = y0; o.y = y1; o.z = y2; o.w = y0 * y0 + y1 * y1 + y2 * y2;
    yp[n] = o;
}

// ------------------- cdist: 1.07 GB streaming output, NT stores -------------
__global__ void k_cdist(const f4* __restrict__ yp, float* __restrict__ dist) {
    int i  = blockIdx.y;
    int j0 = (blockIdx.x * blockDim.x + threadIdx.x) * 4;
    f4 yi = yp[i];
    const f4* pj = yp + j0;
    f4 out;
#pragma unroll
    for (int t = 0; t < 4; ++t) {
        f4 yj = pj[t];
        float d2 = yi.w + yj.w - 2.f * (yi.x * yj.x + yi.y * yj.y + yi.z * yj.z);
        d2 = fmaxf(d2, 0.f);
        out[t] = (d2 > 0.f) ? sqrtf(d2) : 0.f;
    }
    __builtin_nontemporal_store(out, (f4*)(dist + (size_t)i * Nn + j0));
}

// ---------------------------------------------------------------------------
extern "C" void kernel_launch(void* const* d_in, const int* in_sizes, int n_in,
                              void* d_out, int out_size, void* d_ws, size_t ws_size,
                              hipStream_t stream) {
    const float*     x       = (const float*)d_in[0];
    const long long* ei      = (const long long*)d_in[1];
    const float*     W       = (const float*)d_in[2];
    const float*     b       = (const float*)d_in[3];
    const float*     att_src = (const float*)d_in[4];
    const float*     att_dst = (const float*)d_in[5];
    const float*     Wa      = (const float*)d_in[6];
    const float*     ba      = (const float*)d_in[7];
    const float*     W1      = (const float*)d_in[8];
    const float*     b1      = (const float*)d_in[9];
    const float*     W2      = (const float*)d_in[10];
    const float*     b2      = (const float*)d_in[11];

    // Output tuple, flat in return order: dist [N,N] f32, ei [2,E'] i64, alpha [E',2] f32
    float*     dist   = (float*)d_out;
    long long* ei_out = (long long*)(dist + (size_t)Nn * Nn);
    float*     alpha  = (float*)(ei_out + (size_t)2 * EP);

    // Workspace layout (floats) — total ~46.8 MB
    float*    ws    = (float*)d_ws;
    float*    H     = ws;                           // N*256
    float*    agg   = H   + (size_t)Nn * 256;       // N*256
    float*    a_s   = agg + (size_t)Nn * 256;       // N*2
    float*    a_d   = a_s + (size_t)Nn * 2;         // N*2
    unsigned* m_u   = (unsigned*)(a_d + (size_t)Nn * 2); // N*2
    float*    denom = (float*)(m_u + (size_t)Nn * 2);    // N*2
    float*    h2    = denom + (size_t)Nn * 2;       // N*128
    float*    h3    = h2 + (size_t)Nn * 128;        // N*64
    f4*       yp    = (f4*)(h3 + (size_t)Nn * 64);  // N*4

    k_init<<<(Nn * 256 + 255) / 256, 256, 0, stream>>>(agg, m_u, denom);

    // H = x @ W   (no bias: PyG GATConv adds b only after aggregation)
    k_wmma_gemm<false, false, false, false>
        <<<(Nn / 16) * (256 / 16) / 8, 256, 0, stream>>>(x, nullptr, W, nullptr, H,
                                                         Nn, 256, 256);

    k_attn_scores<<<(Nn * 2 + 255) / 256, 256, 0, stream>>>(H, att_src, att_dst, a_s, a_d);

    k_edge1<<<(EP + 255) / 256, 256, 0, stream>>>(ei, a_s, a_d, ei_out, alpha, m_u);
    k_edge2<<<(EP + 255) / 256, 256, 0, stream>>>(ei_out, m_u, alpha, denom);
    k_edge3<<<(EP + 7) / 8, 256, 0, stream>>>(ei_out, denom, H, alpha, agg);

    // h2 = relu( relu(agg + b) @ Wa + ba )
    k_wmma_gemm<true, true, true, true>
        <<<(Nn / 16) * (128 / 16) / 8, 256, 0, stream>>>(agg, b, Wa, ba, h2,
                                                         Nn, 256, 128);
    // h3 = relu( h2 @ W1 + b1 )
    k_wmma_gemm<false, true, false, true>
        <<<(Nn / 16) * (64 / 16) / 8, 256, 0, stream>>>(h2, nullptr, W1, b1, h3,
                                                        Nn, 128, 64);

    k_y<<<(Nn + 255) / 256, 256, 0, stream>>>(h3, W2, b2, yp);

    k_cdist<<<dim3(Nn / 1024, Nn), 256, 0, stream>>>(yp, dist);
}